// GNN_42820823941340
// MI455X (gfx1250) — compile-verified
//
#include <hip/hip_runtime.h>

typedef __attribute__((ext_vector_type(2))) float v2f;
typedef __attribute__((ext_vector_type(8))) float v8f;
typedef __attribute__((ext_vector_type(4))) unsigned int u32x4;
typedef __attribute__((ext_vector_type(4))) int i32x4;
typedef __attribute__((ext_vector_type(8))) int i32x8;

#define NPTS 20
#define DIM  128
#define KNN  5
#define XSTR 132   // LDS row stride for X (floats); 132 % 64 = 4 -> conflict-free fragment loads
#define GSTR 33    // LDS row stride for Gram (floats)
#define WPB  2     // waves (=samples) per block

#if defined(__has_builtin)
#if __has_builtin(__builtin_amdgcn_tensor_load_to_lds) && __has_builtin(__builtin_amdgcn_s_wait_tensorcnt)
#define USE_TDM 1
#endif
#endif

__global__ __launch_bounds__(WPB * 32)
void knn_gram_wmma(const float* __restrict__ x,
                   int* __restrict__ outU,
                   int* __restrict__ outV,
                   int nSamples)
{
    extern __shared__ float smem[];   // [WPB][32*XSTR] X tiles, then [WPB][32*GSTR] Gram

    const int lane = threadIdx.x & 31;
    const int wave = threadIdx.x >> 5;
    const int s = blockIdx.x * WPB + wave;
    if (s >= nSamples) return;   // wave-uniform: EXEC stays all-ones for WMMA

    float* Xl = smem + wave * (32 * XSTR);
    float* Gl = smem + WPB * 32 * XSTR + wave * (32 * GSTR);

    const float* xs = x + (size_t)s * (NPTS * DIM);

    // ---- Stage 1: 20x128 fp32 tile -> LDS with row stride 132 floats ----
#ifdef USE_TDM
    {
        // Tensor Data Mover: one DMA op per wave; hardware inserts the 4-DWORD
        // row pad (pad_interval code 6 = 128 DWORDs, pad_amount code 3 = 4 DWORDs)
        // so rows land at the XSTR=132 conflict-free stride.
        const unsigned ldsOff = (unsigned)((const char*)Xl - (const char*)smem); // dyn-LDS base = 0
        const unsigned long long ga = (unsigned long long)xs;
        u32x4 g0;
        g0.x = 1u;                                         // count=1, is_restore=0, no gather
        g0.y = ldsOff;                                     // lds_addr (bytes)
        g0.z = (unsigned)(ga & 0xFFFFFFFFu);               // global_addr[31:0]
        g0.w = (unsigned)((ga >> 32) & 0x1FFFFFFu)         // global_addr[56:32]
             | (2u << 30);                                 // type=2 ("image")
        i32x8 g1;
        g1[0] = (2 << 16)            // data_size = 4 bytes
              | (1 << 20)            // pad_enable
              | (6 << 22)            // pad_interval: 128 DWORDs
              | (3 << 25);           // pad_amount: 4 DWORDs
        g1[1] = (DIM & 0xFFFF) << 16;      // tensor_dim0[15:0] @ bits 63:48
        g1[2] = (NPTS & 0xFFFF) << 16;     // tensor_dim0[31:16]=0; tensor_dim1[15:0] @ 95:80
        g1[3] = (DIM & 0xFFFF) << 16;      // tensor_dim1[31:16]=0; tile_dim0 @ 127:112
        g1[4] = NPTS;                      // tile_dim1=20; tile_dim2=0 (2D)
        g1[5] = DIM;                       // tensor_dim0_stride[31:0]
        g1[6] = 0;                         // stride[47:32]=0; tensor_dim1_stride[15:0]=0
        g1[7] = 0;
        i32x4 gz4 = {0, 0, 0, 0};                   // groups 2/3 unused (<=2D tensor)
        i32x8 gz8 = {0, 0, 0, 0, 0, 0, 0, 0};       // 6-arg toolchain: extra zero group
        __builtin_amdgcn_tensor_load_to_lds(g0, g1, gz4, gz4, gz8, 0);
    }
#else
    #pragma unroll
    for (int r = 0; r < NPTS; ++r) {
        const float4 v = ((const float4*)(xs + r * DIM))[lane];   // 32 lanes * 16B = full row
        ((float4*)(Xl + r * XSTR))[lane] = v;
    }
#endif
    // zero-pad rows 20..31 (disjoint from the TDM-written region)
    for (int idx = lane; idx < (32 - NPTS) * XSTR; idx += 32)
        Xl[NPTS * XSTR + idx] = 0.0f;

#ifdef USE_TDM
    __builtin_amdgcn_s_wait_tensorcnt((short)0);
    asm volatile("" ::: "memory");   // keep fragment ds_loads below the wait
#endif

    // ---- Stage 2: Gram G = X * X^T with V_WMMA_F32_16X16X4_F32 ----
    // A 16x4 fragment: lane%16 = M, lane/16 selects K pair. B mirrors it with
    // lane%16 = N, so for a Gram the A and B fragments are the same registers:
    // 2 ds_load_b64 per K-step feed all four 16x16 tiles.
    const int lmod = lane & 15;
    const int koff = (lane >> 4) * 2;
    const float* rowLo = Xl + lmod * XSTR;          // rows 0..15 block
    const float* rowHi = Xl + (16 + lmod) * XSTR;   // rows 16..31 block

    v8f acc00 = {}, acc01 = {}, acc10 = {}, acc11 = {};
    #pragma unroll 4
    for (int kb = 0; kb < DIM; kb += 4) {
        v2f r0 = *(const v2f*)(rowLo + kb + koff);
        v2f r1 = *(const v2f*)(rowHi + kb + koff);
        acc00 = __builtin_amdgcn_wmma_f32_16x16x4_f32(false, r0, false, r0,
                                                      (short)0, acc00, false, false);
        acc01 = __builtin_amdgcn_wmma_f32_16x16x4_f32(false, r0, false, r1,
                                                      (short)0, acc01, false, false);
        acc10 = __builtin_amdgcn_wmma_f32_16x16x4_f32(false, r1, false, r0,
                                                      (short)0, acc10, false, false);
        acc11 = __builtin_amdgcn_wmma_f32_16x16x4_f32(false, r1, false, r1,
                                                      (short)0, acc11, false, false);
    }

    // ---- Stage 3: scatter D fragments (lane=N, vgpr v -> M = v + 8*(lane/16)) ----
    const int moff = (lane >> 4) * 8;
    #pragma unroll
    for (int v = 0; v < 8; ++v) {
        Gl[(v + moff) * GSTR + lmod]           = acc00[v];
        Gl[(v + moff) * GSTR + 16 + lmod]      = acc01[v];
        Gl[(16 + v + moff) * GSTR + lmod]      = acc10[v];
        Gl[(16 + v + moff) * GSTR + 16 + lmod] = acc11[v];
    }

    // ---- Stage 4: per-row 5-NN selection, register-resident ----
    // rank(d2(i,j)) = rank(G[j][j] - 2 G[i][j])  (sq_i is constant per row).
    // Strict '<' argmin keeps lowest index on ties (top_k tie-break); self
    // excluded up front (reference drops the distance-0 self entry).
    if (lane < NPTS) {
        float score[NPTS];
        #pragma unroll
        for (int j = 0; j < NPTS; ++j)
            score[j] = Gl[j * GSTR + j] - 2.0f * Gl[lane * GSTR + j];

        unsigned chosen = 1u << lane;
        const long long base = ((long long)s * NPTS + lane) * KNN;
        #pragma unroll
        for (int t = 0; t < KNN; ++t) {
            float bmin = 3.4e38f;
            int bj = 0;
            #pragma unroll
            for (int j = 0; j < NPTS; ++j) {
                const float d = ((chosen >> j) & 1u) ? 3.4e38f : score[j];
                if (d < bmin) { bmin = d; bj = j; }
            }
            chosen |= 1u << bj;
            outU[base + t] = lane;
            outV[base + t] = bj;
        }
    }
}

extern "C" void kernel_launch(void* const* d_in, const int* in_sizes, int n_in,
                              void* d_out, int out_size, void* d_ws, size_t ws_size,
                              hipStream_t stream) {
    const float* x = (const float*)d_in[0];
    const int nSamples = in_sizes[0] / (NPTS * DIM);   // 16384
    int* outU = (int*)d_out;                            // U then V, concatenated flat
    int* outV = outU + (size_t)nSamples * NPTS * KNN;
    const int blocks = (nSamples + WPB - 1) / WPB;
    const size_t shmem = (size_t)WPB * 32 * (XSTR + GSTR) * sizeof(float);
    knn_gram_wmma<<<blocks, WPB * 32, shmem, stream>>>(x, outU, outV, nSamples);
}